// GCNConv_layer_72258529788100
// MI455X (gfx1250) — compile-verified
//
#include <hip/hip_runtime.h>
#include <math.h>

#define N_HIDDEN 128
#define N_GRAPHS 128
#define GN_EPS 1e-5f
#define XS_STRIDE 132  // 16x128 tile padded: 132*4B = 528B row stride, bank-conflict free, 16B aligned

typedef float v2f __attribute__((ext_vector_type(2)));
typedef float v8f __attribute__((ext_vector_type(8)));

// Native fp32 global atomic add (no return -> STOREcnt, resolves at L2, DEV scope).
__device__ __forceinline__ void atomic_add_f32(float* p, float v) {
  asm volatile("global_atomic_add_f32 %0, %1, off scope:SCOPE_DEV"
               :
               : "v"(p), "v"(v)
               : "memory");
}

// ---------------------------------------------------------------------------
// Init: agg <- bias (out = scatter_sum + b), deg <- 1 (self loop weight),
//       per-graph stats <- 0.   One flat grid covers everything.
// ---------------------------------------------------------------------------
__global__ __launch_bounds__(256) void k_init(float* __restrict__ agg,
                                              const float* __restrict__ bias,
                                              float* __restrict__ deg,
                                              float* __restrict__ cnt,
                                              float* __restrict__ gsum,
                                              float* __restrict__ gsumsq,
                                              int nNodes) {
  int i = blockIdx.x * blockDim.x + threadIdx.x;
  int total = nNodes * N_HIDDEN;
  if (i < total) agg[i] = bias[i & (N_HIDDEN - 1)];
  if (i < nNodes) deg[i] = 1.0f;
  if (i < N_GRAPHS * N_HIDDEN) { gsum[i] = 0.0f; gsumsq[i] = 0.0f; }
  if (i < N_GRAPHS) cnt[i] = 0.0f;
}

// ---------------------------------------------------------------------------
// deg[dst] += w  (self-loop "1" already in init)
// ---------------------------------------------------------------------------
__global__ __launch_bounds__(256) void k_degree(const int* __restrict__ dst,
                                                const float* __restrict__ w,
                                                float* __restrict__ deg, int nE) {
  int e = blockIdx.x * blockDim.x + threadIdx.x;
  if (e < nE) atomic_add_f32(&deg[dst[e]], w[e]);
}

__global__ __launch_bounds__(256) void k_dinv(const float* __restrict__ deg,
                                              float* __restrict__ dinv, int nNodes) {
  int n = blockIdx.x * blockDim.x + threadIdx.x;
  if (n < nNodes) {
    float d = deg[n];
    dinv[n] = d > 0.0f ? rsqrtf(d) : 0.0f;
  }
}

// ---------------------------------------------------------------------------
// h = x @ W^T via V_WMMA_F32_16X16X4_F32.
// Block = 256 threads = 8 waves. Wave w owns output columns [16w, 16w+16).
// Each wave preloads its 128(K) x 16(N) slice of B (= W^T) into 32 v2f frags
// (held in VGPRs for the whole kernel), then loops over 16-row tiles of x,
// staging each tile in LDS.
// ---------------------------------------------------------------------------
__global__ __launch_bounds__(256) void k_gemm_wmma(const float* __restrict__ x,
                                                   const float* __restrict__ W,
                                                   float* __restrict__ h,
                                                   int nNodes) {
  __shared__ float xs[16 * XS_STRIDE];
  const int tid   = threadIdx.x;
  const int wave  = tid >> 5;        // 0..7
  const int lane  = tid & 31;
  const int lhalf = lane >> 4;       // 0 or 1
  const int l16   = lane & 15;
  const int colBase = wave * 16;

  // Preload B fragments: B(K, N) = W[N][K]
  v2f bfrag[32];
  {
    const float* wrow = W + (size_t)(colBase + l16) * N_HIDDEN + 2 * lhalf;
#pragma unroll
    for (int kk = 0; kk < 32; ++kk) {
      bfrag[kk].x = wrow[kk * 4 + 0];
      bfrag[kk].y = wrow[kk * 4 + 1];
    }
  }

  const int rowTiles = nNodes >> 4;  // full 16-row tiles
  for (int rt = blockIdx.x; rt < rowTiles; rt += gridDim.x) {
    // Stage the 16x128 x tile: 256 threads x 8 floats (two float4 each)
    {
      const int r     = tid >> 4;        // 0..15
      const int cbase = (tid & 15) * 8;  // 0,8,...,120
      const float4* gp =
          reinterpret_cast<const float4*>(x + ((size_t)(rt * 16 + r)) * N_HIDDEN + cbase);
      float4* sp = reinterpret_cast<float4*>(&xs[r * XS_STRIDE + cbase]);
      sp[0] = gp[0];
      sp[1] = gp[1];
    }
    __syncthreads();

    v8f acc = {};
    const float* arow = &xs[l16 * XS_STRIDE + 2 * lhalf];
#pragma unroll
    for (int kk = 0; kk < 32; ++kk) {
      v2f a;
      a.x = arow[kk * 4 + 0];
      a.y = arow[kk * 4 + 1];
      acc = __builtin_amdgcn_wmma_f32_16x16x4_f32(
          /*neg_a=*/false, a, /*neg_b=*/false, bfrag[kk],
          /*c_mod=*/(short)0, acc, /*reuse_a=*/false, /*reuse_b=*/false);
    }

    // Store D: rows rt*16 + r + 8*lhalf, col = colBase + l16
    float* hp = h + ((size_t)(rt * 16 + 8 * lhalf)) * N_HIDDEN + colBase + l16;
#pragma unroll
    for (int r = 0; r < 8; ++r) hp[(size_t)r * N_HIDDEN] = acc[r];
    __syncthreads();
  }
}

// Scalar fallback for any tail rows (nNodes % 16)
__global__ __launch_bounds__(256) void k_gemm_tail(const float* __restrict__ x,
                                                   const float* __restrict__ W,
                                                   float* __restrict__ h,
                                                   int startRow, int nNodes) {
  int i = blockIdx.x * blockDim.x + threadIdx.x;
  int rows = nNodes - startRow;
  if (i >= rows * N_HIDDEN) return;
  int r = startRow + (i >> 7);
  int c = i & (N_HIDDEN - 1);
  float s = 0.0f;
  for (int k = 0; k < N_HIDDEN; ++k)
    s += x[(size_t)r * N_HIDDEN + k] * W[(size_t)c * N_HIDDEN + k];
  h[(size_t)r * N_HIDDEN + c] = s;
}

// ---------------------------------------------------------------------------
// Edge scatter: agg[dst][:] += dinv[src]*w*dinv[dst] * h[src][:]
// One wave per edge; lane gathers float4 (global_load_b128), 4 native atomics.
// ---------------------------------------------------------------------------
__global__ __launch_bounds__(256) void k_scatter(const float* __restrict__ h,
                                                 const int* __restrict__ src,
                                                 const int* __restrict__ dst,
                                                 const float* __restrict__ w,
                                                 const float* __restrict__ dinv,
                                                 float* __restrict__ agg, int nE) {
  const int wave = threadIdx.x >> 5;
  const int lane = threadIdx.x & 31;
  const int e = blockIdx.x * 8 + wave;
  if (e >= nE) return;
  const int s = src[e];
  const int d = dst[e];
  const float nrm = dinv[s] * w[e] * dinv[d];
  const float4 hv =
      *reinterpret_cast<const float4*>(&h[(size_t)s * N_HIDDEN + lane * 4]);
  float* ap = &agg[(size_t)d * N_HIDDEN + lane * 4];
  atomic_add_f32(ap + 0, nrm * hv.x);
  atomic_add_f32(ap + 1, nrm * hv.y);
  atomic_add_f32(ap + 2, nrm * hv.z);
  atomic_add_f32(ap + 3, nrm * hv.w);
}

// ---------------------------------------------------------------------------
// Fused: add self-loop term (dinv[n]^2 * h[n]) in place, accumulate per-graph
// sum and count. batch is SORTED, so each wave register-accumulates across its
// 16 nodes and only flushes an atomic when the (wave-uniform) graph id changes.
// Block covers 32 consecutive nodes: half = tid>>7 walks nodes n0+half+2k.
// ---------------------------------------------------------------------------
__global__ __launch_bounds__(256) void k_node_stats(const float* __restrict__ h,
                                                    const float* __restrict__ dinv,
                                                    const int* __restrict__ batch,
                                                    float* __restrict__ agg,
                                                    float* __restrict__ gsum,
                                                    float* __restrict__ cnt, int nNodes) {
  const int c = threadIdx.x & (N_HIDDEN - 1);
  const int half = threadIdx.x >> 7;
  const int n0 = blockIdx.x * 32 + half;
  float acc = 0.0f, cacc = 0.0f;
  int curg = -1;
  for (int k = 0; k < 16; ++k) {
    int n = n0 + 2 * k;
    if (n >= nNodes) break;
    int g = batch[n];
    float di = dinv[n];
    size_t idx = (size_t)n * N_HIDDEN + c;
    float y = agg[idx] + di * di * h[idx];
    agg[idx] = y;
    if (g != curg) {  // wave-uniform branch (same node across the wave)
      if (curg >= 0) {
        atomic_add_f32(&gsum[curg * N_HIDDEN + c], acc);
        if (c == 0) atomic_add_f32(&cnt[curg], cacc);
      }
      curg = g; acc = 0.0f; cacc = 0.0f;
    }
    acc += y; cacc += 1.0f;
  }
  if (curg >= 0) {
    atomic_add_f32(&gsum[curg * N_HIDDEN + c], acc);
    if (c == 0) atomic_add_f32(&cnt[curg], cacc);
  }
}

__global__ __launch_bounds__(256) void k_mean(const float* __restrict__ gsum,
                                              const float* __restrict__ cnt,
                                              float* __restrict__ mean) {
  int i = blockIdx.x * blockDim.x + threadIdx.x;
  if (i < N_GRAPHS * N_HIDDEN) mean[i] = gsum[i] / fmaxf(cnt[i >> 7], 1.0f);
}

// Variance pass with the same run-length compressed atomics.
__global__ __launch_bounds__(256) void k_var(const float* __restrict__ agg,
                                             const int* __restrict__ batch,
                                             const float* __restrict__ mean,
                                             const float* __restrict__ mscale,
                                             float* __restrict__ gsumsq, int nNodes) {
  const int c = threadIdx.x & (N_HIDDEN - 1);
  const int half = threadIdx.x >> 7;
  const int n0 = blockIdx.x * 32 + half;
  const float ms = mscale[c];
  float acc = 0.0f;
  int curg = -1;
  float mu = 0.0f;
  for (int k = 0; k < 16; ++k) {
    int n = n0 + 2 * k;
    if (n >= nNodes) break;
    int g = batch[n];
    if (g != curg) {
      if (curg >= 0) atomic_add_f32(&gsumsq[curg * N_HIDDEN + c], acc);
      curg = g; acc = 0.0f;
      mu = ms * mean[g * N_HIDDEN + c];
    }
    float cen = agg[(size_t)n * N_HIDDEN + c] - mu;
    acc += cen * cen;
  }
  if (curg >= 0) atomic_add_f32(&gsumsq[curg * N_HIDDEN + c], acc);
}

__global__ __launch_bounds__(256) void k_invstd(const float* __restrict__ gsumsq,
                                                const float* __restrict__ cnt,
                                                float* __restrict__ invstd) {
  int i = blockIdx.x * blockDim.x + threadIdx.x;
  if (i < N_GRAPHS * N_HIDDEN)
    invstd[i] = rsqrtf(gsumsq[i] / fmaxf(cnt[i >> 7], 1.0f) + GN_EPS);
}

__global__ __launch_bounds__(256) void k_final(const float* __restrict__ agg,
                                               const int* __restrict__ batch,
                                               const float* __restrict__ mean,
                                               const float* __restrict__ invstd,
                                               const float* __restrict__ mscale,
                                               const float* __restrict__ gnw,
                                               const float* __restrict__ gnb,
                                               float* __restrict__ out, int nNodes) {
  int i = blockIdx.x * blockDim.x + threadIdx.x;
  if (i >= nNodes * N_HIDDEN) return;
  int n = i >> 7;
  int c = i & (N_HIDDEN - 1);
  int g = batch[n];
  float cen = agg[i] - mscale[c] * mean[g * N_HIDDEN + c];
  float v = gnw[c] * cen * invstd[g * N_HIDDEN + c] + gnb[c];
  out[i] = fmaxf(v, 0.0f);
}

// ---------------------------------------------------------------------------
extern "C" void kernel_launch(void* const* d_in, const int* in_sizes, int n_in,
                              void* d_out, int out_size, void* d_ws, size_t ws_size,
                              hipStream_t stream) {
  const float* node  = (const float*)d_in[0];
  const int*   ei    = (const int*)d_in[1];  // [2, E]
  const float* eattr = (const float*)d_in[2];
  const int*   batch = (const int*)d_in[3];
  const float* W     = (const float*)d_in[4];
  const float* bias  = (const float*)d_in[5];
  const float* gnw   = (const float*)d_in[6];
  const float* gnb   = (const float*)d_in[7];
  const float* gms   = (const float*)d_in[8];

  const int nNodes = in_sizes[0] / N_HIDDEN;
  const int nE     = in_sizes[2];
  const int* srcIdx = ei;
  const int* dstIdx = ei + nE;

  float* ws = (float*)d_ws;
  size_t off = 0;
  float* h      = ws + off; off += (size_t)nNodes * N_HIDDEN;
  float* agg    = ws + off; off += (size_t)nNodes * N_HIDDEN;
  float* deg    = ws + off; off += (size_t)nNodes;
  float* dinv   = ws + off; off += (size_t)nNodes;
  float* cnt    = ws + off; off += N_GRAPHS;
  float* gsum   = ws + off; off += N_GRAPHS * N_HIDDEN;
  float* gsumsq = ws + off; off += N_GRAPHS * N_HIDDEN;
  float* mean   = ws + off; off += N_GRAPHS * N_HIDDEN;
  float* invstd = ws + off; off += N_GRAPHS * N_HIDDEN;

  float* out = (float*)d_out;

  const int total    = nNodes * N_HIDDEN;
  const int gFlat    = (total + 255) / 256;
  const int gEdges   = (nE + 255) / 256;
  const int gNodes   = (nNodes + 255) / 256;
  const int gWaveE   = (nE + 7) / 8;        // 1 wave per edge, 8 edges/block
  const int gChunk   = (nNodes + 31) / 32;  // 32 nodes per block (stats passes)
  const int rowTiles = nNodes >> 4;

  k_init<<<gFlat, 256, 0, stream>>>(agg, bias, deg, cnt, gsum, gsumsq, nNodes);

  if (rowTiles > 0)
    k_gemm_wmma<<<rowTiles, 256, 0, stream>>>(node, W, h, nNodes);
  if (nNodes & 15) {
    int startRow = rowTiles << 4;
    int tail = (nNodes - startRow) * N_HIDDEN;
    k_gemm_tail<<<(tail + 255) / 256, 256, 0, stream>>>(node, W, h, startRow, nNodes);
  }

  k_degree<<<gEdges, 256, 0, stream>>>(dstIdx, eattr, deg, nE);
  k_dinv<<<gNodes, 256, 0, stream>>>(deg, dinv, nNodes);
  k_scatter<<<gWaveE, 256, 0, stream>>>(h, srcIdx, dstIdx, eattr, dinv, agg, nE);
  k_node_stats<<<gChunk, 256, 0, stream>>>(h, dinv, batch, agg, gsum, cnt, nNodes);
  k_mean<<<(N_GRAPHS * N_HIDDEN + 255) / 256, 256, 0, stream>>>(gsum, cnt, mean);
  k_var<<<gChunk, 256, 0, stream>>>(agg, batch, mean, gms, gsumsq, nNodes);
  k_invstd<<<(N_GRAPHS * N_HIDDEN + 255) / 256, 256, 0, stream>>>(gsumsq, cnt, invstd);
  k_final<<<gFlat, 256, 0, stream>>>(agg, batch, mean, invstd, gms, gnw, gnb, out, nNodes);
}